// AConnect_5978594476054
// MI455X (gfx1250) — compile-verified
//
#include <hip/hip_runtime.h>
#include <hip/hip_bf16.h>
#include <stdint.h>

// Problem constants (match reference)
#define BATCH 256
#define IN_F  512
#define OUT_F 512

// Tiling: each block = one batch sample x 128 output columns.
// Werr streamed through LDS with the Tensor Data Mover, double buffered.
#define COLS   128            // output columns per block
#define ROWS   32             // Werr rows (i-dim) per TDM chunk
#define NCHUNK (IN_F / ROWS)  // 16 chunks

#if defined(__gfx1250__)
typedef unsigned int u32x4 __attribute__((ext_vector_type(4)));
typedef int          i32x8 __attribute__((ext_vector_type(8)));
typedef int          i32x4 __attribute__((ext_vector_type(4)));

// Build a TDM descriptor for a 2D tile (tile_rows x COLS f32 elements) of a
// 512x512 f32 tensor with row stride 512 elements, and issue the DMA.
// D# layout per CDNA5 ISA ch.8 (group0 128b, group1 256b; groups 2/3 unused->0).
// Toolchain here is the 6-arg clang-23 form:
//   (u32x4 g0, i32x8 g1, i32x4 g2, i32x4 g3, i32x8 extra, i32 cpol)
__device__ __forceinline__ void tdm_load_tile_2d(uint32_t lds_addr,
                                                 uint64_t gaddr,
                                                 int tile_rows)
{
    u32x4 g0;
    g0[0] = 1u;                                        // count=1 (valid user D#)
    g0[1] = lds_addr;                                  // LDS byte address of tile
    g0[2] = (uint32_t)(gaddr & 0xFFFFFFFFull);         // global_addr[31:0]
    g0[3] = (uint32_t)((gaddr >> 32) & 0x01FFFFFFull)  // global_addr[56:32]
          | (2u << 30);                                // type = 2 ("image")

    i32x8 g1;
    g1[0] = (int)(2u << 16);              // data_size = 2 -> 4-byte elements
    g1[1] = (int)((uint32_t)OUT_F << 16); // tensor_dim0 = 512 (bits 79:48, low16)
    g1[2] = (int)((uint32_t)IN_F  << 16); // tensor_dim1 = 512 (bits 111:80, low16)
    g1[3] = (int)((uint32_t)COLS  << 16); // tile_dim0 = 128   (bits 127:112)
    g1[4] = tile_rows;                    // tile_dim1; tile_dim2 = 0
    g1[5] = OUT_F;                        // tensor_dim0_stride = 512 (low 32 of 48)
    g1[6] = 0;                            // stride hi16 | tensor_dim1_stride lo16
    g1[7] = 0;                            // tensor_dim1_stride hi32 (unused, 2D tile)

    i32x4 gz4 = (i32x4){0, 0, 0, 0};            // groups 2/3: zero (no dims 2..4)
    i32x8 gz8 = (i32x8){0, 0, 0, 0, 0, 0, 0, 0};
    __builtin_amdgcn_tensor_load_to_lds(g0, g1, gz4, gz4, gz8, 0);
}
#endif

__global__ void __launch_bounds__(COLS)
aconnect_stream_kernel(const float* __restrict__ X,
                       const float* __restrict__ W,
                       const float* __restrict__ bias,
                       const float* __restrict__ Werr,
                       const float* __restrict__ Berr,
                       const int*   __restrict__ idx,
                       float*       __restrict__ out)
{
    __shared__ __align__(16) float sX[IN_F];
    __shared__ __align__(16) float sE[2][ROWS * COLS];

    const int t  = threadIdx.x;          // 0..127
    const int b  = blockIdx.y;           // batch sample
    const int o0 = blockIdx.x * COLS;    // output-column tile base
    const int o  = o0 + 4 * t;           // this thread's 4 columns

    // Stage X[b, :] into LDS: 128 threads x float4 = 512 floats.
    *(float4*)&sX[4 * t] = *(const float4*)&X[(size_t)b * IN_F + 4 * t];

    const int id = idx[b];
    const uint64_t eBase = (uint64_t)(size_t)Werr
        + 4ull * ((uint64_t)id * (uint64_t)IN_F * OUT_F + (uint64_t)o0);

    float4 acc = make_float4(0.f, 0.f, 0.f, 0.f);

#if defined(__gfx1250__)
    // Prologue: kick chunk 0 DMA (wave 0 only; TDM is a wave-level op).
    if (t < 32)
        tdm_load_tile_2d((uint32_t)(size_t)&sE[0][0], eBase, ROWS);

    for (int c = 0; c < NCHUNK; ++c) {
        const int cur = c & 1;
        if (t < 32) {
            if (c + 1 < NCHUNK) {
                // Issue next chunk into the other buffer, then wait until the
                // oldest (chunk c) has landed: TDM ops are in-order per wave,
                // so tensorcnt<=1 means chunk c is complete.
                tdm_load_tile_2d((uint32_t)(size_t)&sE[cur ^ 1][0],
                                 eBase + (uint64_t)(c + 1) * (ROWS * OUT_F * 4ull),
                                 ROWS);
                __builtin_amdgcn_s_wait_tensorcnt(1);
            } else {
                __builtin_amdgcn_s_wait_tensorcnt(0);
            }
        }
        __syncthreads();                  // sE[cur] (and sX on c==0) visible

        const int i0 = c * ROWS;
        #pragma unroll 8
        for (int r = 0; r < ROWS; ++r) {
            const float  xs = sX[i0 + r];
            const float4 w  = *(const float4*)&W[(i0 + r) * OUT_F + o];
            const float4 e  = *(const float4*)&sE[cur][r * COLS + 4 * t];
            acc.x = fmaf(xs * w.x, e.x, acc.x);
            acc.y = fmaf(xs * w.y, e.y, acc.y);
            acc.z = fmaf(xs * w.z, e.z, acc.z);
            acc.w = fmaf(xs * w.w, e.w, acc.w);
        }
        __syncthreads();                  // all waves done reading sE[cur]
    }
#else
    // Host pass / non-gfx1250 fallback: synchronous cooperative staging.
    for (int c = 0; c < NCHUNK; ++c) {
        __syncthreads();
        for (int k = t; k < ROWS * COLS; k += COLS) {
            const int rr = k / COLS, cc = k % COLS;
            sE[0][k] = Werr[(size_t)id * IN_F * OUT_F
                            + (size_t)(c * ROWS + rr) * OUT_F + o0 + cc];
        }
        __syncthreads();
        const int i0 = c * ROWS;
        for (int r = 0; r < ROWS; ++r) {
            const float  xs = sX[i0 + r];
            const float4 w  = *(const float4*)&W[(i0 + r) * OUT_F + o];
            const float4 e  = *(const float4*)&sE[0][r * COLS + 4 * t];
            acc.x = fmaf(xs * w.x, e.x, acc.x);
            acc.y = fmaf(xs * w.y, e.y, acc.y);
            acc.z = fmaf(xs * w.z, e.z, acc.z);
            acc.w = fmaf(xs * w.w, e.w, acc.w);
        }
    }
#endif

    // Epilogue: Z += bias * Berr[idx[b]]
    const float4 bv = *(const float4*)&bias[o];
    const float4 be = *(const float4*)&Berr[(size_t)id * OUT_F + o];
    float4 r;
    r.x = acc.x + bv.x * be.x;
    r.y = acc.y + bv.y * be.y;
    r.z = acc.z + bv.z * be.z;
    r.w = acc.w + bv.w * be.w;
    *(float4*)&out[(size_t)b * OUT_F + o] = r;
}

extern "C" void kernel_launch(void* const* d_in, const int* in_sizes, int n_in,
                              void* d_out, int out_size, void* d_ws, size_t ws_size,
                              hipStream_t stream) {
    (void)in_sizes; (void)n_in; (void)out_size; (void)d_ws; (void)ws_size;
    const float* X    = (const float*)d_in[0];
    const float* W    = (const float*)d_in[1];
    const float* bias = (const float*)d_in[2];
    const float* Werr = (const float*)d_in[3];
    const float* Berr = (const float*)d_in[4];
    const int*   idx  = (const int*)d_in[5];
    float*       out  = (float*)d_out;

    dim3 grid(OUT_F / COLS, BATCH);   // (4, 256)
    dim3 block(COLS);                 // 128 threads = 4 waves
    hipLaunchKernelGGL(aconnect_stream_kernel, grid, block, 0, stream,
                       X, W, bias, Werr, Berr, idx, out);
}